// ASTRF_53919019434059
// MI455X (gfx1250) — compile-verified
//
#include <hip/hip_runtime.h>

typedef __attribute__((ext_vector_type(2))) float v2f;
typedef __attribute__((ext_vector_type(4))) float v4f;
typedef __attribute__((ext_vector_type(8))) float v8f;

#define BATCH   4
#define IN_DIM  8
#define OUT_DIM 64
#define N_WIN   64
#define N_SEQ   4096
#define N_REAL  6144
#define KTOT    (IN_DIM * N_WIN)   // 512
#define KCHUNK  128
#define LWSTR   132                // padded weight row stride (floats) -> bank spread
#define TT      64                 // output t-tile per workgroup
#define XSTR    128
#define NTILES  (N_REAL / TT)      // 96

__device__ __forceinline__ int lower_bound_i32(const int* __restrict__ a, int n, int v) {
    int lo = 0, hi = n;
    while (lo < hi) {
        int m = (lo + hi) >> 1;
        if (a[m] < v) lo = m + 1; else hi = m;
    }
    return lo;
}

__launch_bounds__(256)
__global__ void astrf_fused(const float* __restrict__ x,      // (B, IN_DIM, N_SEQ)
                            const float* __restrict__ weight, // (O, IN_DIM, N_WIN) == Wmat[o, k=i*64+w]
                            const float* __restrict__ bias,   // (O)
                            const int*   __restrict__ srcIdx, // (B, N_SEQ) sorted unique
                            float*       __restrict__ out) {  // (B, O, N_REAL)
    __shared__ __align__(16) float lw[OUT_DIM * LWSTR]; // 33,792 B weight K-chunk
    __shared__ __align__(16) float xs[IN_DIM * XSTR];   //  4,096 B scattered stimulus (reversed cols)

    const int tid  = threadIdx.x;
    const int bb   = blockIdx.x / NTILES;
    const int tile = blockIdx.x % NTILES;
    const int t0   = tile * TT;

    // ---- build reversed XS[i, c] with c = t0 + 64 - idx, idx in [t0-63, t0+63] -> c in [1,127]
    for (int j = tid; j < IN_DIM * XSTR; j += 256) xs[j] = 0.0f;

    const int* idxb = srcIdx + bb * N_SEQ;
    const int sLo = lower_bound_i32(idxb, N_SEQ, t0 - 63);
    const int sHi = lower_bound_i32(idxb, N_SEQ, t0 + TT);
    __syncthreads();
    for (int s = sLo + tid; s < sHi; s += 256) {
        int c = t0 + 64 - idxb[s];
        #pragma unroll
        for (int i = 0; i < IN_DIM; ++i)
            xs[i * XSTR + c] = x[(bb * IN_DIM + i) * N_SEQ + s];
    }

    // ---- wave -> output-tile mapping (16 tiles of 16x16, 8 waves x 2 tiles)
    const int lane  = tid & 31;
    const int wv    = tid >> 5;   // 0..7
    const int mT    = wv & 3;     // o tile
    const int nP    = wv >> 2;    // 0..1 -> n tiles 2*nP, 2*nP+1
    const int laneM = lane & 15;
    const int laneH = lane >> 4;

    v8f acc0 = {};
    v8f acc1 = {};

    const int tb0  = nP * 32 + laneM;               // N coord of first tile
    const int aRow = (mT * 16 + laneM) * LWSTR;

    for (int kc = 0; kc < KTOT; kc += KCHUNK) {
        __syncthreads();
        // cooperative, coalesced load of weight chunk Wmat[:, kc..kc+127] -> LDS
        for (int v = tid; v < OUT_DIM * (KCHUNK / 4); v += 256) {
            int o  = v >> 5;
            int j4 = (v & 31) << 2;
            v4f w4 = *(const v4f*)(weight + o * KTOT + kc + j4);
            *(v4f*)(&lw[o * LWSTR + j4]) = w4;
        }
        __syncthreads();

        #pragma unroll 4
        for (int kk = 0; kk < KCHUNK; kk += 4) {
            const int kA = kk + 2 * laneH;          // even: lanes<16 hold K,K+1; lanes>=16 hold K+2,K+3
            // A fragment: aligned 8B LDS load (aRow + kA is even)
            v2f a = *(const v2f*)(&lw[aRow + kA]);

            const int kg = kc + kA;                 // global k -> (i = k>>6, w = k&63)
            const int i0 = kg >> 6;
            const int w0 = kg & 63;
            // kA even => (k, k+1) never crosses the w=63 boundary: same i, w ascends

            // im2col B fragments from reversed XS: col(k) = 64 - t + w, ascends with k
            const int base = i0 * XSTR + 64 - tb0 + w0;
            v2f b0, b1;
            b0.x = xs[base];
            b0.y = xs[base + 1];
            b1.x = xs[base - 16];                   // second tile: t + 16
            b1.y = xs[base - 15];

            acc0 = __builtin_amdgcn_wmma_f32_16x16x4_f32(false, a, false, b0,
                                                         (short)0, acc0, false, false);
            acc1 = __builtin_amdgcn_wmma_f32_16x16x4_f32(false, a, false, b1,
                                                         (short)0, acc1, false, false);
        }
    }

    // ---- epilogue: D[lane][vd] = C[M = vd + 8*laneH, N = laneM]; add bias, store
    const int oBase = mT * 16 + 8 * laneH;
    const int tA    = t0 + tb0;
    #pragma unroll
    for (int vd = 0; vd < 8; ++vd) {
        const int o  = oBase + vd;
        const float bv = bias[o];
        float* op = out + (bb * OUT_DIM + o) * N_REAL;
        op[tA]      = acc0[vd] + bv;
        op[tA + 16] = acc1[vd] + bv;
    }
}

extern "C" void kernel_launch(void* const* d_in, const int* in_sizes, int n_in,
                              void* d_out, int out_size, void* d_ws, size_t ws_size,
                              hipStream_t stream) {
    (void)in_sizes; (void)n_in; (void)out_size; (void)d_ws; (void)ws_size;
    const float* x      = (const float*)d_in[0];
    const float* weight = (const float*)d_in[1];
    const float* bias   = (const float*)d_in[2];
    const int*   srcIdx = (const int*)d_in[3];
    float*       out    = (float*)d_out;

    dim3 grid(BATCH * NTILES);   // 384 workgroups, 256 threads (8 wave32) each
    astrf_fused<<<grid, 256, 0, stream>>>(x, weight, bias, srcIdx, out);
}